// RON_9156870275320
// MI455X (gfx1250) — compile-verified
//
#include <hip/hip_runtime.h>
#include <hip/hip_bf16.h>

#define MBS    16384
#define IN_DIM 512
#define H_DIM  512
#define NC     10
#define T_STEPS 10

typedef __attribute__((ext_vector_type(2))) float v2f;
typedef __attribute__((ext_vector_type(8))) float v8f;

// ---------------------------------------------------------------------------
// Fused WMMA f32 GEMM + oscillator update.
//   acc[m,n] = sum_k Yin[m,k] * BT[n*512 + k]      (BT holds B^T, k-contiguous)
// mode 0: out_a0[m,n] = acc + bias[n]                       (precompute x@W0^T+b0)
// mode 1: g = acc + a0[m,n];        z/y oscillator update   (layer 0)
// mode 2: g = acc + bias[n] + (yo@W2)[m,n]; z/y update      (layer 1, K ext = NC)
// Wave tile: 32(M) x 64(N)  = 2 M-subtiles x 4 N-subtiles, k-step 4 via
// V_WMMA_F32_16X16X4_F32; B fragments are reused across both M-subtiles
// (0.75 loads per WMMA). Block: 256 thr = 8 waves as 4(M) x 2(N) ->
// 128 x 128 block tile. Grid: (MBS/128, H/128).
// ---------------------------------------------------------------------------
__global__ __launch_bounds__(256) void ron_gemm_fused(
    const float* __restrict__ Yin,  const float* __restrict__ BT,
    const float* __restrict__ bias, const float* __restrict__ a0,
    const float* __restrict__ yo,   const float* __restrict__ W2,
    const float* __restrict__ gamma,const float* __restrict__ eps,
    const float* __restrict__ y_old, float* __restrict__ z,
    float* __restrict__ y_new, float* __restrict__ out_a0, int mode)
{
    const int tid   = threadIdx.x;
    const int wave  = tid >> 5;
    const int lane  = tid & 31;
    const int lm    = lane & 15;          // M-index (A) / N-index (B,C)
    const int lk    = (lane >> 4) << 1;   // k-pair base within k-step (0 or 2)
    const int waveM = wave & 3;
    const int waveN = wave >> 2;
    const int mBase = blockIdx.x * 128 + waveM * 32;
    const int nBase = blockIdx.y * 128 + waveN * 64;

    v8f acc[2][4];
    #pragma unroll
    for (int i = 0; i < 2; ++i)
        #pragma unroll
        for (int t = 0; t < 4; ++t) acc[i][t] = (v8f){0,0,0,0,0,0,0,0};

    const float* arow0 = Yin + (size_t)(mBase      + lm) * 512;
    const float* arow1 = Yin + (size_t)(mBase + 16 + lm) * 512;
    const float* bptr[4];
    #pragma unroll
    for (int t = 0; t < 4; ++t)
        bptr[t] = BT + (size_t)(nBase + t * 16 + lm) * 512;

    for (int k0 = 0; k0 < 512; k0 += 4) {
        const v2f a0f = *(const v2f*)(arow0 + k0 + lk);
        const v2f a1f = *(const v2f*)(arow1 + k0 + lk);
        #pragma unroll
        for (int t = 0; t < 4; ++t) {
            const v2f b = *(const v2f*)(bptr[t] + k0 + lk);
            acc[0][t] = __builtin_amdgcn_wmma_f32_16x16x4_f32(
                false, a0f, false, b, (short)0, acc[0][t], false, false);
            acc[1][t] = __builtin_amdgcn_wmma_f32_16x16x4_f32(
                false, a1f, false, b, (short)0, acc[1][t], false, false);
        }
    }

    if (mode == 2) {
        // Fold yo @ W2  ([MBS,NC] x [NC,H]) into the accumulator: K=10 pad to 12.
        const float* yrow0 = yo + (size_t)(mBase      + lm) * NC;
        const float* yrow1 = yo + (size_t)(mBase + 16 + lm) * NC;
        #pragma unroll
        for (int k0 = 0; k0 < 12; k0 += 4) {
            const int ka = k0 + lk;
            v2f a0f, a1f;
            a0f.x = (ka     < NC) ? yrow0[ka]     : 0.0f;
            a0f.y = (ka + 1 < NC) ? yrow0[ka + 1] : 0.0f;
            a1f.x = (ka     < NC) ? yrow1[ka]     : 0.0f;
            a1f.y = (ka + 1 < NC) ? yrow1[ka + 1] : 0.0f;
            #pragma unroll
            for (int t = 0; t < 4; ++t) {
                const int n = nBase + t * 16 + lm;
                v2f b;
                b.x = (ka     < NC) ? W2[(size_t)ka       * H_DIM + n] : 0.0f;
                b.y = (ka + 1 < NC) ? W2[(size_t)(ka + 1) * H_DIM + n] : 0.0f;
                acc[0][t] = __builtin_amdgcn_wmma_f32_16x16x4_f32(
                    false, a0f, false, b, (short)0, acc[0][t], false, false);
                acc[1][t] = __builtin_amdgcn_wmma_f32_16x16x4_f32(
                    false, a1f, false, b, (short)0, acc[1][t], false, false);
            }
        }
    }

    // Epilogue. C/D layout: VGPR j -> M = j + 8*(lane>=16), N = lane%16.
    #pragma unroll
    for (int i = 0; i < 2; ++i) {
        const int mOff = mBase + i * 16 + ((lane >> 4) << 3);
        #pragma unroll
        for (int t = 0; t < 4; ++t) {
            const int n = nBase + t * 16 + lm;
            if (mode == 0) {
                const float bb = bias[n];
                #pragma unroll
                for (int j = 0; j < 8; ++j)
                    out_a0[(size_t)(mOff + j) * H_DIM + n] = acc[i][t][j] + bb;
            } else {
                const float gadd = (mode == 2) ? bias[n] : 0.0f;
                const float gm   = gamma[n];
                const float om   = 1.0f - eps[n];
                #pragma unroll
                for (int j = 0; j < 8; ++j) {
                    const size_t idx = (size_t)(mOff + j) * H_DIM + n;
                    float g  = acc[i][t][j] + gadd + ((mode == 1) ? a0[idx] : 0.0f);
                    float zo = z[idx];
                    float yv = y_old[idx];
                    float zn = tanhf(g) + zo * om - gm * yv;   // tau = 1
                    z[idx]     = zn;
                    y_new[idx] = yv + zn;                      // y + tau*z_new
                }
            }
        }
    }
}

// yout_new = tanh(y1 @ W2^T + b2)   [MBS, NC] — tiny, plain VALU dot products.
__global__ __launch_bounds__(256) void ron_out(
    const float* __restrict__ y1, const float* __restrict__ W2,
    const float* __restrict__ b2, float* __restrict__ yo_new)
{
    const int idx = blockIdx.x * 256 + threadIdx.x;
    const int m = idx >> 4, c = idx & 15;
    if (c >= NC) return;
    const float* yr = y1 + (size_t)m * H_DIM;
    const float* wr = W2 + (size_t)c * H_DIM;
    float s = b2[c];
    for (int k = 0; k < H_DIM; k += 4) {
        float4 a = *(const float4*)(yr + k);
        float4 w = *(const float4*)(wr + k);
        s += a.x * w.x + a.y * w.y + a.z * w.z + a.w * w.w;
    }
    yo_new[(size_t)m * NC + c] = tanhf(s);
}

// W1T[n*512 + k] = W1[k*512 + n]  (one-time, 1 MB)
__global__ __launch_bounds__(256) void transpose512(
    const float* __restrict__ W, float* __restrict__ WT)
{
    const int idx = blockIdx.x * 256 + threadIdx.x;
    const int r = idx >> 9, c = idx & 511;
    WT[(size_t)c * 512 + r] = W[(size_t)r * 512 + c];
}

extern "C" void kernel_launch(void* const* d_in, const int* in_sizes, int n_in,
                              void* d_out, int out_size, void* d_ws, size_t ws_size,
                              hipStream_t stream) {
    (void)in_sizes; (void)n_in; (void)out_size; (void)ws_size;

    const float* x    = (const float*)d_in[0];
    const float* z0i  = (const float*)d_in[1];
    const float* z1i  = (const float*)d_in[2];
    const float* y0i  = (const float*)d_in[3];
    const float* y1i  = (const float*)d_in[4];
    const float* yoi  = (const float*)d_in[5];
    const float* W0   = (const float*)d_in[6];
    const float* b0   = (const float*)d_in[7];
    const float* W1   = (const float*)d_in[8];
    const float* b1   = (const float*)d_in[9];
    const float* W2   = (const float*)d_in[10];
    const float* b2   = (const float*)d_in[11];
    const float* gam  = (const float*)d_in[12];
    const float* eps  = (const float*)d_in[13];
    // d_in[14] = labels (unused at beta=0); d_in[15] = T (fixed at 10 by setup)

    const size_t MH = (size_t)MBS * H_DIM;
    float* o_z0 = (float*)d_out;
    float* o_z1 = o_z0 + MH;
    float* o_y0 = o_z1 + MH;
    float* o_y1 = o_y0 + MH;
    float* o_yo = o_y1 + MH;

    float* ws    = (float*)d_ws;
    float* a0    = ws;  ws += MH;                       // x@W0^T + b0 (loop-invariant)
    float* W1T   = ws;  ws += (size_t)H_DIM * H_DIM;    // W1 transposed
    float* y0alt = ws;  ws += MH;                       // ping-pong partners
    float* y1alt = ws;  ws += MH;
    float* yoalt = ws;  /* MBS*NC */

    // Initialize states (inputs must not be mutated; T even -> parity-0 = d_out).
    hipMemcpyAsync(o_z0, z0i, MH * sizeof(float), hipMemcpyDeviceToDevice, stream);
    hipMemcpyAsync(o_z1, z1i, MH * sizeof(float), hipMemcpyDeviceToDevice, stream);
    hipMemcpyAsync(o_y0, y0i, MH * sizeof(float), hipMemcpyDeviceToDevice, stream);
    hipMemcpyAsync(o_y1, y1i, MH * sizeof(float), hipMemcpyDeviceToDevice, stream);
    hipMemcpyAsync(o_yo, yoi, (size_t)MBS * NC * sizeof(float), hipMemcpyDeviceToDevice, stream);

    transpose512<<<dim3((H_DIM * H_DIM) / 256), 256, 0, stream>>>(W1, W1T);

    const dim3 ggrid(MBS / 128, H_DIM / 128);
    // a0 = x @ W0^T + b0 : B[k][n] = W0[n*IN + k] -> BT = W0 directly.
    ron_gemm_fused<<<ggrid, 256, 0, stream>>>(x, W0, b0, nullptr, nullptr, nullptr,
                                              nullptr, nullptr, nullptr, nullptr,
                                              nullptr, a0, 0);

    float* y0buf[2] = {o_y0, y0alt};
    float* y1buf[2] = {o_y1, y1alt};
    float* yobuf[2] = {o_yo, yoalt};

    for (int t = 0; t < T_STEPS; ++t) {
        const int p = t & 1, q = p ^ 1;
        // Layer 0: g0 = a0 + y1@W1 ; B[k][n] = W1[k*H+n] = W1T[n*H+k] -> BT = W1T.
        ron_gemm_fused<<<ggrid, 256, 0, stream>>>(y1buf[p], W1T, nullptr, a0,
                                                  nullptr, nullptr, gam, eps,
                                                  y0buf[p], o_z0, y0buf[q], nullptr, 1);
        // Layer 1: g1 = y0@W1^T + b1 + yo@W2 ; B[k][n] = W1[n*H+k] -> BT = W1.
        ron_gemm_fused<<<ggrid, 256, 0, stream>>>(y0buf[p], W1, b1, nullptr,
                                                  yobuf[p], W2, gam, eps,
                                                  y1buf[p], o_z1, y1buf[q], nullptr, 2);
        // Output layer: yout = tanh(y1@W2^T + b2).
        ron_out<<<dim3((MBS * 16) / 256), 256, 0, stream>>>(y1buf[p], W2, b2, yobuf[q]);
    }
    // After an even number of steps the final z/y/yout states live in d_out.
}